// CRFLayer_57964878627499
// MI455X (gfx1250) — compile-verified
//
#include <hip/hip_runtime.h>
#include <stdint.h>

#define Bq 2048
#define Tq 512
#define Kq 5
#define TCH 32               // timesteps per TDM chunk
#define NCH (Tq / TCH)       // 16 chunks

typedef unsigned int v4u __attribute__((ext_vector_type(4)));
typedef int          v8i __attribute__((ext_vector_type(8)));
typedef int          v4i __attribute__((ext_vector_type(4)));

// ---- Tensor Data Mover: 3-D tile load  Global(B,T,K) -> LDS tile laid out [t][b][k] ----
// dim0 (x): K=5 dwords contiguous
// dim1 (y): 32 batch rows, stride T*K dwords
// dim2 (z): TCH timesteps, stride K dwords
// LDS fill order is x,y,z-sequential -> layout [z][y][x] = [t][b][k],
// lane stride = 5 dwords -> conflict-free on 64 banks.
__device__ __forceinline__ void tdm_issue_tile(const float* gptr, unsigned lds_addr) {
  unsigned long long ga = (unsigned long long)(uintptr_t)gptr;
  v4u g0;
  g0[0] = 1u;                                   // count=1, user descriptor
  g0[1] = lds_addr;                             // lds_addr (bytes)
  g0[2] = (unsigned)(ga & 0xFFFFFFFFu);         // global_addr[31:0]
  g0[3] = (unsigned)((ga >> 32) & 0x01FFFFFFu)  // global_addr[56:32]
        | 0x80000000u;                          // type=2 (bits 127:126)

  const unsigned td0 = Kq, td1 = 32u, tl0 = Kq, tl1 = 32u, tl2 = TCH;
  const unsigned long long s0 = (unsigned long long)Tq * Kq;   // 2560 elems
  const unsigned long long s1 = (unsigned long long)Kq;        // 5 elems
  v8i g1;
  g1[0] = (int)(2u << 16);                                     // data_size=4B, no pad/iterate, wg_mask=0
  g1[1] = (int)((td0 & 0xFFFFu) << 16);                        // tensor_dim0[15:0] @bit48
  g1[2] = (int)((td0 >> 16) | ((td1 & 0xFFFFu) << 16));        // tensor_dim0 hi | tensor_dim1 lo
  g1[3] = (int)((td1 >> 16) | (tl0 << 16));                    // tensor_dim1 hi | tile_dim0
  g1[4] = (int)(tl1 | (tl2 << 16));                            // tile_dim1 | tile_dim2
  g1[5] = (int)(unsigned)(s0 & 0xFFFFFFFFu);                   // tensor_dim0_stride[31:0]
  g1[6] = (int)((unsigned)((s0 >> 32) & 0xFFFFu)
        | (unsigned)((s1 & 0xFFFFu) << 16));                   // stride0 hi | stride1 lo
  g1[7] = (int)(unsigned)((s1 >> 16) & 0xFFFFFFFFu);           // stride1 hi
  v4i g2 = { (int)TCH, 0, 0, 0 };                              // tensor_dim2 (z bound)
  v4i g3 = { 0, 0, 0, 0 };
  v8i g4 = { 0, 0, 0, 0, 0, 0, 0, 0 };                         // 6-arg form extension
  __builtin_amdgcn_tensor_load_to_lds(g0, g1, g2, g3, g4, 0);
}

__device__ __forceinline__ float sel5(const float e[Kq], int i) {
  float r = e[0];
  r = (i == 1) ? e[1] : r;
  r = (i == 2) ? e[2] : r;
  r = (i == 3) ? e[3] : r;
  r = (i == 4) ? e[4] : r;
  return r;
}

__device__ __forceinline__ int argmax5(const float v[Kq]) {
  float bv = v[0]; int ba = 0;
#pragma unroll
  for (int k = 1; k < Kq; ++k) { if (v[k] > bv) { bv = v[k]; ba = k; } }
  return ba;   // strict '>' keeps first max == jnp.argmax tie rule
}

__global__ __launch_bounds__(32) void crf_scan_kernel(
    const float* __restrict__ inputs,   // (B,T,K) f32
    const float* __restrict__ trans,    // (K,K) f32
    const int*   __restrict__ lengths,  // (B,)
    const int*   __restrict__ labels,   // (B,T)
    float*       __restrict__ out,      // best_path (B,T) as float, then loss
    unsigned short* __restrict__ bp,    // (T,B) packed backpointers (5x3 bits)
    float*       __restrict__ bsum)     // per-block loss partials
{
  __shared__ float tile[2][TCH * 32 * Kq];   // ~41 KB double buffer
  __shared__ float ltr[32];                  // transition copy for dynamic index

  const int lane = threadIdx.x;
  const int b    = blockIdx.x * 32 + lane;

  if (lane < Kq * Kq) ltr[lane] = trans[lane];
  __syncthreads();

  // Uniform transition -> SGPR-resident for the unrolled recurrences.
  float trr[Kq][Kq];
#pragma unroll
  for (int j = 0; j < Kq; ++j)
#pragma unroll
    for (int k = 0; k < Kq; ++k) trr[j][k] = trans[j * Kq + k];

  const int len = lengths[b];
  const int* lab = labels + (size_t)b * Tq;

  float alpha[Kq], vp[Kq];
  float gold = 0.f;
  int ptr0 = 0, prevl = 0;

  const float* gbase = inputs + (size_t)(blockIdx.x * 32) * Tq * Kq;

  // Prime the pipeline: chunk 0 -> buffer 0
  tdm_issue_tile(gbase, (unsigned)(uintptr_t)(const void*)&tile[0][0]);

  for (int c = 0; c < NCH; ++c) {
    if (c + 1 < NCH) {
      tdm_issue_tile(gbase + (size_t)(c + 1) * TCH * Kq,
                     (unsigned)(uintptr_t)(const void*)&tile[(c + 1) & 1][0]);
      __builtin_amdgcn_s_wait_tensorcnt(1);   // chunk c landed (in-order)
    } else {
      __builtin_amdgcn_s_wait_tensorcnt(0);
    }
    asm volatile("" ::: "memory");
    const float* tb = &tile[c & 1][0];
    asm volatile("" : "+v"(tb));              // launder: TDM wrote this LDS

    for (int tl = 0; tl < TCH; ++tl) {
      const int t = c * TCH + tl;
      const float* ep = tb + (tl * 32 + lane) * Kq;
      float e[Kq];
#pragma unroll
      for (int k = 0; k < Kq; ++k) e[k] = ep[k];

      if (t == 0) {
        // scores[0,b,j,k] = e0[k] for every j:
        //   LSE over j of 5 equal values = e0[k] + ln(5); max = e0[k], arg=0.
#pragma unroll
        for (int k = 0; k < Kq; ++k) {
          alpha[k] = e[k] + 1.6094379124341003f;  // ln(5)
          vp[k]    = e[k];
        }
        const int l0 = lab[0];
        gold += sel5(e, l0);
        prevl = l0;
        if (len == 1) ptr0 = argmax5(vp);
      } else if (t < len) {
        const int lt = lab[t];
        float na[Kq], nv[Kq];
        unsigned pk = 0u;
#pragma unroll
        for (int k = 0; k < Kq; ++k) {
          // forward: alpha'[k] = e[k] + LSE_j(alpha[j] + T[j][k])
          const float c0 = alpha[0] + trr[0][k];
          const float c1 = alpha[1] + trr[1][k];
          const float c2 = alpha[2] + trr[2][k];
          const float c3 = alpha[3] + trr[3][k];
          const float c4 = alpha[4] + trr[4][k];
          const float mm = fmaxf(fmaxf(fmaxf(c0, c1), fmaxf(c2, c3)), c4);
          const float sm = __expf(c0 - mm) + __expf(c1 - mm) + __expf(c2 - mm)
                         + __expf(c3 - mm) + __expf(c4 - mm);
          na[k] = e[k] + mm + __logf(sm);
          // viterbi: vp'[k] = e[k] + max_j(vp[j] + T[j][k]), record argmax
          float bv = vp[0] + trr[0][k]; int ba = 0;
#pragma unroll
          for (int j = 1; j < Kq; ++j) {
            const float cv = vp[j] + trr[j][k];
            if (cv > bv) { bv = cv; ba = j; }
          }
          nv[k] = e[k] + bv;
          pk |= (unsigned)ba << (3 * k);
        }
#pragma unroll
        for (int k = 0; k < Kq; ++k) { alpha[k] = na[k]; vp[k] = nv[k]; }
        gold += sel5(e, lt) + ltr[prevl * Kq + lt];
        prevl = lt;
        bp[(size_t)t * Bq + b] = (unsigned short)pk;   // coalesced across lanes
        if (t == len - 1) ptr0 = argmax5(vp);
      }
      // t >= len: alpha/vp frozen (masked_scatter semantics), bp[t] never read
    }
  }

  // ---- loss contribution: LSE_k(alpha) - gold, deterministic wave reduce ----
  float mm = alpha[0];
#pragma unroll
  for (int k = 1; k < Kq; ++k) mm = fmaxf(mm, alpha[k]);
  float sm = 0.f;
#pragma unroll
  for (int k = 0; k < Kq; ++k) sm += __expf(alpha[k] - mm);
  float v = mm + __logf(sm) - gold;
#pragma unroll
  for (int off = 16; off > 0; off >>= 1) v += __shfl_down(v, off, 32);
  if (lane == 0) bsum[blockIdx.x] = v;

  // ---- backtrace (each lane reads only its own stored column) ----
  int p = ptr0;
  out[(size_t)b * Tq + (Tq - 1)] = (float)p;
  for (int t = Tq - 2; t >= 0; --t) {
    const int tn = t + 1;
    if (tn < len) {
      const unsigned pk = bp[(size_t)tn * Bq + b];
      p = (int)((pk >> (3 * p)) & 7u);
    }
    out[(size_t)b * Tq + t] = (float)p;
  }
}

__global__ void crf_finalize_kernel(const float* __restrict__ bsum,
                                    float* __restrict__ out) {
  if (threadIdx.x == 0 && blockIdx.x == 0) {
    float s = 0.f;
    for (int i = 0; i < Bq / 32; ++i) s += bsum[i];   // fixed order: deterministic
    out[(size_t)Bq * Tq] = s / (float)Bq;
  }
}

extern "C" void kernel_launch(void* const* d_in, const int* in_sizes, int n_in,
                              void* d_out, int out_size, void* d_ws, size_t ws_size,
                              hipStream_t stream) {
  (void)in_sizes; (void)n_in; (void)out_size; (void)ws_size;
  const float* inputs  = (const float*)d_in[0];   // (B,T,K) f32
  const float* trans   = (const float*)d_in[1];   // (K,K)   f32
  /* d_in[2] = mask (bool) — redundant with lengths, unused */
  const int*   lengths = (const int*)d_in[3];     // (B,)
  const int*   labels  = (const int*)d_in[4];     // (B,T)
  float* out = (float*)d_out;

  unsigned short* bp = (unsigned short*)d_ws;                       // T*B*2 = 2 MB
  float* bsum = (float*)((char*)d_ws + (size_t)Tq * Bq * sizeof(unsigned short));

  crf_scan_kernel<<<Bq / 32, 32, 0, stream>>>(inputs, trans, lengths, labels,
                                              out, bp, bsum);
  crf_finalize_kernel<<<1, 32, 0, stream>>>(bsum, out);
}